// HPO_TSN_FPHA_net_85074712199852
// MI455X (gfx1250) — compile-verified
//
#include <hip/hip_runtime.h>
#include <hip/hip_bf16.h>

// Problem constants (fixed by the reference setup)
#define BS    2560          // clip_batch(512) * segments(5)
#define CH    320           // 64 * D channels
#define HW    169           // 13*13 spatial
#define SLAB  (CH * HW)     // 54080 floats per sample
#define K63   63            // uvd channels (21 joints * 3)
#define KPAD  64
#define DDEP  5
#define CELLS (HW * DDEP)   // 845 cells per (b,k)
#define NCLS  45
#define NPAD  48
#define SEG   5
#define GROUPS (BS / SEG)   // 512

typedef __attribute__((ext_vector_type(2))) float v2f;
typedef __attribute__((ext_vector_type(8))) float v8f;

__device__ __forceinline__ float sigmoidf(float x) {
    return 1.0f / (1.0f + __expf(-x));
}

// ---------------------------------------------------------------------------
// Kernel 1: (b, 64, D, H, W) -> (b, 63, H, W, D) permute, sigmoid on k<3.
// For fixed (b,k) the 5 source rows are a contiguous 845-float span, so both
// the global load and the global store are fully coalesced; the transpose
// happens through LDS. This is the bandwidth-dominant kernel (~1.1 GB total).
// Scalar b32 on the global side: the 845-float spans are only 4B aligned for
// odd k, so vector b64/b128 accesses would be misaligned.
// ---------------------------------------------------------------------------
__global__ void permute_uvd_kernel(const float* __restrict__ in,
                                   float* __restrict__ out1) {
    __shared__ float slab[CELLS];
    const int k = blockIdx.x;   // 0..62
    const int b = blockIdx.y;   // 0..2559

    const float* src = in + (size_t)b * SLAB + (size_t)k * CELLS;
    for (int i = threadIdx.x; i < CELLS; i += blockDim.x)
        slab[i] = src[i];
    __syncthreads();

    float* dst = out1 + (size_t)b * ((size_t)K63 * CELLS) + (size_t)k * CELLS;
    const bool sig = (k < 3);   // joint 0 (hand root) gets sigmoid
    for (int o = threadIdx.x; o < CELLS; o += blockDim.x) {
        const int d  = o % DDEP;
        const int hw = o / DDEP;
        float v = slab[d * HW + hw];
        if (sig) v = sigmoidf(v);
        dst[o] = v;
    }
}

// ---------------------------------------------------------------------------
// Kernel 2: per-sample argmax over 845 conf cells (sigmoid is monotonic ->
// argmax on raw values), first-occurrence tie-break to match jax top_k.
// Then gather the 63 uvd values of the winning cell (K padded to 64).
// ---------------------------------------------------------------------------
__global__ void argmax_best_kernel(const float* __restrict__ in,
                                   float* __restrict__ bestbuf,  // (BS, 64)
                                   float* __restrict__ top_out)  // (BS)
{
    __shared__ float slab[CELLS];
    __shared__ float rv[256];
    __shared__ int   rp[256];
    const int b   = blockIdx.x;
    const int tid = threadIdx.x;

    // conf channels 315..319 -> contiguous 845-float span
    const float* conf = in + (size_t)b * SLAB + (size_t)(K63 * DDEP) * HW;
    for (int i = tid; i < CELLS; i += blockDim.x)
        slab[i] = conf[i];
    __syncthreads();

    float bv = -3.402823466e38f;
    int   bp = 0x7fffffff;
    for (int p = tid; p < CELLS; p += blockDim.x) {
        const float v = slab[(p % DDEP) * HW + (p / DDEP)];
        if (v > bv) { bv = v; bp = p; }   // strict > keeps lowest p per thread
    }
    rv[tid] = bv;
    rp[tid] = bp;
    __syncthreads();

    for (int s = 128; s > 0; s >>= 1) {
        if (tid < s) {
            const float ov = rv[tid + s];
            const int   op = rp[tid + s];
            if (ov > rv[tid] || (ov == rv[tid] && op < rp[tid])) {
                rv[tid] = ov;
                rp[tid] = op;
            }
        }
        __syncthreads();
    }

    const int best_p = rp[0];
    const int bd  = best_p % DDEP;
    const int bhw = best_p / DDEP;

    if (tid < KPAD) {
        const int k = tid;
        float v = 0.0f;                               // zero-pad K 63 -> 64
        if (k < K63) {
            v = in[(size_t)b * SLAB + (size_t)(k * DDEP + bd) * HW + bhw];
            if (k < 3) v = sigmoidf(v);
        }
        bestbuf[(size_t)b * KPAD + k] = v;
    }
    if (tid == 0)
        top_out[b] = (float)best_p;   // int32 index emitted into fp32 buffer
}

// ---------------------------------------------------------------------------
// Kernel 3a: pad fc weights/bias once so the WMMA loop has zero masking.
// bpair[k2][n] = (B[2*k2][n], B[2*k2+1][n]) with B = fc_w^T zero-padded to
// K=64, N=48; fcb padded to 48.
// ---------------------------------------------------------------------------
__global__ void pad_fc_kernel(const float* __restrict__ fcw,   // (45, 63)
                              const float* __restrict__ fcb,   // (45)
                              float2* __restrict__ bpair,      // (32, 48)
                              float* __restrict__ fcbpad)      // (48)
{
    const int i = blockIdx.x * blockDim.x + threadIdx.x;
    if (i < (KPAD / 2) * NPAD) {
        const int k2 = i / NPAD;
        const int n  = i % NPAD;
        const int k  = 2 * k2;
        float x = (n < NCLS && k < K63)       ? fcw[n * K63 + k]     : 0.0f;
        float y = (n < NCLS && (k + 1) < K63) ? fcw[n * K63 + k + 1] : 0.0f;
        bpair[i] = make_float2(x, y);
    }
    if (i < NPAD)
        fcbpad[i] = (i < NCLS) ? fcb[i] : 0.0f;
}

// ---------------------------------------------------------------------------
// Kernel 3b: logits = best(2560x64pad) @ Bpad(64x48) + bias via
// V_WMMA_F32_16X16X4_F32. One wave per 16x16 tile: 160 M-tiles x 3 N-tiles,
// 16 chained K=4 steps. All loads unconditional: A pairs are contiguous
// (global_load_b64), B pairs come from the padded K-pair buffer
// (lane-coalesced global_load_b64). EXEC stays all-1s throughout.
// A layout: lane holds M = lane%16; VGPR pair holds K {kk,kk+1} (lanes 0-15)
// or K {kk+2,kk+3} (lanes 16-31). B mirrors this for rows of B = fc_w^T.
// ---------------------------------------------------------------------------
__global__ void wmma_gemm_kernel(const float* __restrict__ best,    // (BS, 64)
                                 const float2* __restrict__ bpair,  // (32, 48)
                                 const float* __restrict__ fcbpad,  // (48)
                                 float* __restrict__ logits)        // (BS, 48)
{
    const int tile = blockIdx.x;          // 0..479
    const int tM = tile / 3;
    const int tN = tile % 3;
    const int lane = threadIdx.x & 31;
    const int half = lane >> 4;           // 0: K{+0,+1}, 1: K{+2,+3}
    const int lm   = lane & 15;
    const int M = tM * 16 + lm;           // A row for this lane
    const int N = tN * 16 + lm;           // B column for this lane

    const float* arow = best + (size_t)M * KPAD;
    v8f c = {0.f, 0.f, 0.f, 0.f, 0.f, 0.f, 0.f, 0.f};

#pragma unroll
    for (int kk = 0; kk < KPAD; kk += 4) {
        const int ka = kk + half * 2;     // always even
        v2f a;
        a.x = arow[ka];
        a.y = arow[ka + 1];
        const float2 bv2 = bpair[(ka >> 1) * NPAD + N];
        v2f bb;
        bb.x = bv2.x;
        bb.y = bv2.y;
        c = __builtin_amdgcn_wmma_f32_16x16x4_f32(
                /*neg_a=*/false, a, /*neg_b=*/false, bb,
                /*c_mod=*/(short)0, c, /*reuse_a=*/false, /*reuse_b=*/false);
    }

    const float bias = fcbpad[N];
#pragma unroll
    for (int v = 0; v < 8; ++v) {
        // C/D layout: VGPR v holds M = v (lanes 0-15) or M = 8+v (lanes 16-31)
        const int Mrow = tM * 16 + v + half * 8;
        const int Ncol = tN * 16 + lm;
        logits[(size_t)Mrow * NPAD + Ncol] = c[v] + bias;
    }
}

// ---------------------------------------------------------------------------
// Kernel 4: consensus mean over 5 segments -> (512, 45)
// ---------------------------------------------------------------------------
__global__ void seg_mean_kernel(const float* __restrict__ logits,
                                float* __restrict__ out3) {
    const int i = blockIdx.x * blockDim.x + threadIdx.x;
    if (i >= GROUPS * NCLS) return;
    const int g = i / NCLS;
    const int n = i % NCLS;
    float s = 0.0f;
#pragma unroll
    for (int q = 0; q < SEG; ++q)
        s += logits[(size_t)(g * SEG + q) * NPAD + n];
    out3[i] = s * (1.0f / SEG);
}

extern "C" void kernel_launch(void* const* d_in, const int* in_sizes, int n_in,
                              void* d_out, int out_size, void* d_ws, size_t ws_size,
                              hipStream_t stream) {
    const float* base = (const float*)d_in[0];   // (2560, 320, 13, 13)
    const float* fcw  = (const float*)d_in[1];   // (45, 63)
    const float* fcb  = (const float*)d_in[2];   // (45)

    // Outputs concatenated flat in return order:
    float* out1 = (float*)d_out;                            // pred_uvd_out: BS*63*845
    float* out2 = out1 + (size_t)BS * K63 * CELLS;          // top_idx (as float): BS
    float* out3 = out2 + BS;                                // consensus logits: 512*45

    // Workspace layout (floats):
    //   [0, BS*64)                       bestbuf
    //   [BS*64, BS*64 + BS*48)           logits
    //   [.., +32*48*2)                   bpair (float2, 8B aligned)
    //   [.., +48)                        fcbpad
    float* bestbuf = (float*)d_ws;
    float* logits  = bestbuf + (size_t)BS * KPAD;
    float2* bpair  = (float2*)(logits + (size_t)BS * NPAD);
    float* fcbpad  = (float*)(bpair + (KPAD / 2) * NPAD);

    pad_fc_kernel<<<((KPAD / 2) * NPAD + 255) / 256, 256, 0, stream>>>(
        fcw, fcb, bpair, fcbpad);

    dim3 gPerm(K63, BS);
    permute_uvd_kernel<<<gPerm, 256, 0, stream>>>(base, out1);
    argmax_best_kernel<<<BS, 256, 0, stream>>>(base, bestbuf, out2);
    wmma_gemm_kernel<<<(BS / 16) * 3, 32, 0, stream>>>(bestbuf, bpair, fcbpad, logits);
    seg_mean_kernel<<<(GROUPS * NCLS + 255) / 256, 256, 0, stream>>>(logits, out3);
}